// NKATv9QuantumHamiltonian_32530082300546
// MI455X (gfx1250) — compile-verified
//
#include <hip/hip_runtime.h>

#define DIM 500
#define NB  64

typedef double        v2d  __attribute__((ext_vector_type(2)));
typedef unsigned int  u32x4 __attribute__((ext_vector_type(4)));
typedef int           i32x4 __attribute__((ext_vector_type(4)));
typedef int           i32x8 __attribute__((ext_vector_type(8)));

__constant__ int PRIMES[50] = {
    2, 3, 5, 7, 11, 13, 17, 19, 23, 29,
    31, 37, 41, 43, 47, 53, 59, 61, 67, 71,
    73, 79, 83, 89, 97, 101, 103, 107, 109, 113,
    127, 131, 137, 139, 149, 151, 157, 163, 167, 173,
    179, 181, 191, 193, 197, 199, 211, 223, 227, 229};

// ---------------------------------------------------------------------------
// Kernel 2 (placed first so the disasm snippet shows its body): one block per
// row i. TDM-load C[i,:] (8000 B) into LDS via the Tensor Data Mover, compute
// the 64 per-batch diagonal values exp(-s*ln(i+1)) in fp64, then stream the 64
// output rows with 16-byte non-temporal stores. j is the outer loop so each
// thread reads its C element from LDS exactly once and reuses it for all 64
// batches (64x less LDS traffic than b-outer).
// ---------------------------------------------------------------------------
__global__ void assemble_kernel(const double* __restrict__ C,
                                const float* __restrict__ s_re,
                                const float* __restrict__ s_im,
                                double* __restrict__ out) {
    __shared__ double crow[2 * DIM];   // interleaved (re, im) row of C: 8000 B
    __shared__ double diag_re[NB];
    __shared__ double diag_im[NB];

    const int i   = blockIdx.x;
    const int tid = threadIdx.x;

    // --- TDM: DMA the C row into LDS (wave 0 only; TDM ignores EXEC so the
    //     guard is at whole-wave branch granularity). ------------------------
    if (tid < 32) {
        const unsigned lds_addr = (unsigned)(size_t)(&crow[0]);
        const unsigned long long ga =
            (unsigned long long)(size_t)(C + (size_t)i * (2 * DIM));

        // D# group 0: count=1 | lds_addr | global_addr | type=2 ("image")
        u32x4 g0;
        g0.x = 1u;
        g0.y = lds_addr;
        g0.z = (unsigned)(ga & 0xFFFFFFFFull);
        g0.w = (unsigned)((ga >> 32) & 0x1FFFFFFull) | (2u << 30);

        // D# group 1: data_size=8B, 1D tensor/tile of 1000 elements
        const unsigned td0 = 2u * DIM;          // 1000 x 8-byte elements
        i32x8 g1;
        g1[0] = (int)(3u << 16);                          // data_size = 3 (8B)
        g1[1] = (int)((td0 & 0xFFFFu) << 16);             // tensor_dim0 lo
        g1[2] = (int)((td0 >> 16) | (1u << 16));          // tensor_dim0 hi | tensor_dim1=1
        g1[3] = (int)((td0 & 0xFFFFu) << 16);             // tile_dim0 = 1000
        g1[4] = 0;                                        // tile_dim1/2 unused
        g1[5] = (int)td0;                                 // tensor_dim0_stride lo
        g1[6] = 0;
        g1[7] = 0;

        i32x4 z4 = {0, 0, 0, 0};
#if __clang_major__ >= 23
        i32x8 z8 = {0, 0, 0, 0, 0, 0, 0, 0};
        __builtin_amdgcn_tensor_load_to_lds(g0, g1, z4, z4, z8, 0);
#else
        __builtin_amdgcn_tensor_load_to_lds(g0, g1, z4, z4, 0);
#endif
        __builtin_amdgcn_s_wait_tensorcnt(0);
    }

    // --- per-batch diagonal term: n^{-s} = exp(-sr*L) * (cos(-si*L) + j sin) -
    if (tid < NB) {
        const double L   = log((double)(i + 1));
        const double mag = exp(-(double)s_re[tid] * L);
        const double ang = -(double)s_im[tid] * L;
        diag_re[tid] = mag * cos(ang);
        diag_im[tid] = mag * sin(ang);
    }
    __syncthreads();

    // --- stream 64 output rows. Write-once buffer (256 MB total > 192 MB L2,
    //     never re-read) -> non-temporal b128 stores. ------------------------
    const size_t batch_stride = (size_t)DIM * (size_t)(2 * DIM);  // one matrix
    for (int j = tid; j < DIM; j += blockDim.x) {
        const double cre = crow[2 * j];
        const double cim = crow[2 * j + 1];
        const bool on_diag = (j == i);
        double* p = out + ((size_t)i * (2 * DIM) + 2 * j);
        for (int b = 0; b < NB; ++b, p += batch_stride) {
            v2d v;
            v.x = cre;
            v.y = cim;
            if (on_diag) { v.x += diag_re[b]; v.y += diag_im[b]; }
            __builtin_nontemporal_store(v, (v2d*)p);
        }
    }
}

// ---------------------------------------------------------------------------
// Kernel 1: build the s-independent correction matrix C (complex128, gather
// style so writes are conflict-free). C is stored interleaved (re, im) in d_ws.
// ---------------------------------------------------------------------------
__global__ void build_corr_kernel(double* __restrict__ C) {
    const int idx = blockIdx.x * blockDim.x + threadIdx.x;
    if (idx >= DIM * DIM) return;
    const int i = idx / DIM;
    const int j = idx % DIM;
    const int d = j - i;

    const double theta = 1e-30;
    const double kappa = 1e-20;
    const double es    = 1e-25;

    double re = 0.0, im = 0.0;

    // prime-indexed corrections live only on d in {0, +2, -2}
    if (d == 0 || d == 2 || d == -2) {
        for (int k = 0; k < 50; ++k) {
            const int p = PRIMES[k];                       // p <= 229 < DIM-2 always
            const double a  = theta * log((double)p);      // Re(corr)
            const double bi = a * log((double)(k + 1));    // Im(corr)
            if (d == 0 && i == p - 1) { re += a;            im += bi; }
            else if (d == 2  && i == p - 1) { re += -0.1 * bi; im += 0.1 * a; }  // corr*0.1j
            else if (d == -2 && j == p - 1) { re +=  0.1 * bi; im += 0.1 * a; }  // conj(corr)*0.1j
        }
        // M-theory corrections (n = row+1 <= 100)
        if (d == 0 && i < 100) {
            const double n = (double)(i + 1);
            re += kappa * exp(log(n) / 11.0) * exp(-n / 1000.0);
        }
        if (d == 2 && i < 100 && i < DIM - 3) {
            const double n = (double)(i + 1);
            re += 0.01 * kappa * exp(log(n) / 11.0) * exp(-n / 1000.0);
        }
        if (d == -2 && j < 100 && j < DIM - 3) {
            const double n = (double)(j + 1);
            re += 0.01 * kappa * exp(log(n) / 11.0) * exp(-n / 1000.0);
        }
    }
    // entanglement band: i<30, 1 <= j-i <= 9 -> e*(1+1j); mirror -> e*(1-1j)
    if (i < 30 && d > 0 && d < 10) {
        const double e = es * exp(-(double)d / 5.0);
        re += e; im += e;
    }
    if (j < 30 && d < 0 && d > -10) {
        const double e = es * exp((double)d / 5.0);
        re += e; im -= e;
    }

    C[2 * (size_t)idx]     = re;
    C[2 * (size_t)idx + 1] = im;
}

extern "C" void kernel_launch(void* const* d_in, const int* in_sizes, int n_in,
                              void* d_out, int out_size, void* d_ws, size_t ws_size,
                              hipStream_t stream) {
    const float* s_re = (const float*)d_in[0];
    const float* s_im = (const float*)d_in[1];
    double* C   = (double*)d_ws;    // DIM*DIM complex128 = 4,000,000 bytes
    double* out = (double*)d_out;   // [B, DIM, DIM] complex128, interleaved

    const int total = DIM * DIM;
    build_corr_kernel<<<(total + 255) / 256, 256, 0, stream>>>(C);
    assemble_kernel<<<DIM, 256, 0, stream>>>(C, s_re, s_im, out);
}